// StateEncoder_40312563040655
// MI455X (gfx1250) — compile-verified
//
#include <hip/hip_runtime.h>
#include <hip/hip_bf16.h>

// ---------------------------------------------------------------------------
// StateEncoder for MI455X (gfx1250, wave32).
//   Pass 1: zero partial buffer (hipMemsetAsync, capture-safe)
//   Pass 2: streaming masked segment-sum over node_tokens (512 MB @ 23.3 TB/s
//           => ~22us floor). Sorted segment ids => register accumulation with
//           atomic flush only on segment boundaries. Metadata via SGPR loads,
//           token stream via b128 loads + global_prefetch_b8.
//   Pass 3: mean + question + step_emb + LayerNorm via V_WMMA_F32_16X16X4_F32
//           (exact f32): row sums of X and X^2 as X * ones.
// ---------------------------------------------------------------------------

typedef __attribute__((ext_vector_type(2))) float v2f;
typedef __attribute__((ext_vector_type(8))) float v8f;

__device__ __forceinline__ void flush_seg(float* __restrict__ sums,
                                          float* __restrict__ counts,
                                          int seg, int lane,
                                          float4& acc, float& cnt) {
    float* p = sums + (long)seg * 128 + lane * 4;
    __hip_atomic_fetch_add(p + 0, acc.x, __ATOMIC_RELAXED, __HIP_MEMORY_SCOPE_AGENT);
    __hip_atomic_fetch_add(p + 1, acc.y, __ATOMIC_RELAXED, __HIP_MEMORY_SCOPE_AGENT);
    __hip_atomic_fetch_add(p + 2, acc.z, __ATOMIC_RELAXED, __HIP_MEMORY_SCOPE_AGENT);
    __hip_atomic_fetch_add(p + 3, acc.w, __ATOMIC_RELAXED, __HIP_MEMORY_SCOPE_AGENT);
    if (lane == 0)
        __hip_atomic_fetch_add(counts + seg, cnt, __ATOMIC_RELAXED, __HIP_MEMORY_SCOPE_AGENT);
    acc.x = acc.y = acc.z = acc.w = 0.f;
    cnt = 0.f;
}

// One wave (32 lanes) owns `nodes_per_wave` consecutive nodes; lane l holds
// features [4l, 4l+4) => one full 128-float row per wave iteration (512B).
__global__ __launch_bounds__(256) void segsum_kernel(
    const float* __restrict__ node_tokens,
    const int* __restrict__ node_batch,
    const unsigned char* __restrict__ active,   // jax bool => 1 byte/elem
    float* __restrict__ sums,
    float* __restrict__ counts,
    int nodes_per_wave, int N)
{
    const int lane = threadIdx.x & 31;
    int wave = (int)((blockIdx.x * blockDim.x + threadIdx.x) >> 5);
    wave = __builtin_amdgcn_readfirstlane(wave);   // force SGPR => s_load metadata
    const long base = (long)wave * nodes_per_wave;
    if (base >= N) return;

    const float4* __restrict__ tok = (const float4*)node_tokens;

    float4 acc = {0.f, 0.f, 0.f, 0.f};
    float  cnt = 0.f;
    int curseg = node_batch[base];

    // 4-deep software pipeline on the b128 token stream
    long idx = base * 32 + lane;
    float4 r0 = tok[idx], r1 = tok[idx + 32], r2 = tok[idx + 64], r3 = tok[idx + 96];

    const int G = nodes_per_wave >> 2;
    for (int gi = 0; gi < G; ++gi) {
        const long i = base + 4 * gi;

        float4 n0 = {}, n1 = {}, n2 = {}, n3 = {};
        if (gi + 1 < G) {
            const long idn = (i + 4) * 32 + lane;
            n0 = tok[idn]; n1 = tok[idn + 32]; n2 = tok[idn + 64]; n3 = tok[idn + 96];
        }
        if (i + 36 <= N)  // pull the stream into L2 ~16KB ahead
            __builtin_prefetch(node_tokens + (i + 32) * 128 + lane * 4, 0, 1);

        // wave-uniform metadata (SGPR path)
        const int4 s4 = *(const int4*)(node_batch + i);
        const unsigned int a4 = *(const unsigned int*)(active + i);

        if (s4.x != curseg) { flush_seg(sums, counts, curseg, lane, acc, cnt); curseg = s4.x; }
        if (a4 & 0x000000ffu) { acc.x += r0.x; acc.y += r0.y; acc.z += r0.z; acc.w += r0.w; cnt += 1.f; }

        if (s4.y != curseg) { flush_seg(sums, counts, curseg, lane, acc, cnt); curseg = s4.y; }
        if (a4 & 0x0000ff00u) { acc.x += r1.x; acc.y += r1.y; acc.z += r1.z; acc.w += r1.w; cnt += 1.f; }

        if (s4.z != curseg) { flush_seg(sums, counts, curseg, lane, acc, cnt); curseg = s4.z; }
        if (a4 & 0x00ff0000u) { acc.x += r2.x; acc.y += r2.y; acc.z += r2.z; acc.w += r2.w; cnt += 1.f; }

        if (s4.w != curseg) { flush_seg(sums, counts, curseg, lane, acc, cnt); curseg = s4.w; }
        if (a4 & 0xff000000u) { acc.x += r3.x; acc.y += r3.y; acc.z += r3.z; acc.w += r3.w; cnt += 1.f; }

        r0 = n0; r1 = n1; r2 = n2; r3 = n3;
    }
    flush_seg(sums, counts, curseg, lane, acc, cnt);
}

// One wave handles 16 graphs (a 16x128 tile). Row sums of X and X.^2 via
// V_WMMA_F32_16X16X4_F32 with B = ones (exact f32 FMA chain).
// A-layout (16x4 f32): lane r (<16): K = {4c, 4c+1}; lane r+16: K = {4c+2, 4c+3}.
__global__ __launch_bounds__(32) void finalize_kernel(
    const float* __restrict__ sums,
    const float* __restrict__ counts,
    const float* __restrict__ question,
    const float* __restrict__ step_weight,
    const int* __restrict__ step_counts,
    const float* __restrict__ ln_gamma,
    const float* __restrict__ ln_beta,
    float* __restrict__ out,
    int max_steps)
{
    const int lane = threadIdx.x;       // 0..31
    const int r    = lane & 15;         // tile row
    const int hi   = lane >> 4;         // K-half select
    const int g    = blockIdx.x * 16 + r;

    const float cnt = counts[g];
    const float inv = 1.0f / fmaxf(cnt, 1.0f);
    int st  = step_counts[g];
    st = st < 0 ? 0 : (st > max_steps ? max_steps : st);
    const int rem = max_steps - st;     // in [0, max_steps]

    const float* __restrict__ sm = sums        + (long)g   * 128;
    const float* __restrict__ qu = question    + (long)g   * 128;
    const float* __restrict__ se = step_weight + (long)rem * 128;

    float xs[64];
    v8f accS = {};   // row sums of x
    v8f accQ = {};   // row sums of x^2
    const v2f ones = {1.0f, 1.0f};

#pragma unroll
    for (int c = 0; c < 32; ++c) {
        const int k0 = 4 * c + 2 * hi;
        const float2 s = *(const float2*)(sm + k0);
        const float2 q = *(const float2*)(qu + k0);
        const float2 e = *(const float2*)(se + k0);
        v2f a;
        a.x = fmaf(s.x, inv, q.x + e.x);
        a.y = fmaf(s.y, inv, q.y + e.y);
        xs[2 * c]     = a.x;
        xs[2 * c + 1] = a.y;
        accS = __builtin_amdgcn_wmma_f32_16x16x4_f32(
            false, a, false, ones, (short)0, accS, false, false);
        v2f a2; a2.x = a.x * a.x; a2.y = a.y * a.y;
        accQ = __builtin_amdgcn_wmma_f32_16x16x4_f32(
            false, a2, false, ones, (short)0, accQ, false, false);
    }

    // D layout: lanes 0-15 hold rows 0-7 in v[0..7]; lanes 16-31 hold rows 8-15.
    // Columns are replicated, so lanes 0 and 16 publish mu / rstd via LDS.
    __shared__ float s_mu[16];
    __shared__ float s_rs[16];
    if (r == 0) {
#pragma unroll
        for (int j = 0; j < 8; ++j) {
            const int row = hi * 8 + j;
            const float mu  = accS[j] * (1.0f / 128.0f);
            const float var = accQ[j] * (1.0f / 128.0f) - mu * mu;
            s_mu[row] = mu;
            s_rs[row] = __frsqrt_rn(var + 1e-5f);
        }
    }
    __syncthreads();
    const float mu = s_mu[r];
    const float rs = s_rs[r];

    float* __restrict__ o = out + (long)g * 128;
#pragma unroll
    for (int c = 0; c < 32; ++c) {
        const int k0 = 4 * c + 2 * hi;
        const float2 gm = *(const float2*)(ln_gamma + k0);
        const float2 bt = *(const float2*)(ln_beta + k0);
        float2 y;
        y.x = fmaf((xs[2 * c]     - mu) * rs, gm.x, bt.x);
        y.y = fmaf((xs[2 * c + 1] - mu) * rs, gm.y, bt.y);
        *(float2*)(o + k0) = y;
    }
}

extern "C" void kernel_launch(void* const* d_in, const int* in_sizes, int n_in,
                              void* d_out, int out_size, void* d_ws, size_t ws_size,
                              hipStream_t stream) {
    const float*         node_tokens = (const float*)d_in[0];
    const float*         question    = (const float*)d_in[1];
    const float*         step_weight = (const float*)d_in[2];
    const float*         ln_gamma    = (const float*)d_in[3];
    const float*         ln_beta     = (const float*)d_in[4];
    const int*           node_batch  = (const int*)d_in[5];
    const unsigned char* active      = (const unsigned char*)d_in[6];
    const int*           step_counts = (const int*)d_in[7];

    const int H = 128;
    const int N = in_sizes[0] / H;
    const int B = in_sizes[1] / H;
    const int max_steps = in_sizes[2] / H - 1;

    float* sums   = (float*)d_ws;                  // [B,128]
    float* counts = sums + (size_t)B * H;          // [B]

    hipMemsetAsync(d_ws, 0, ((size_t)B * H + B) * sizeof(float), stream);

    const int NPW = 128;                           // nodes per wave
    const int waves  = (N + NPW - 1) / NPW;        // 8192 waves for N=1M
    const int blocks = (waves + 7) / 8;            // 8 waves / 256-thread block
    segsum_kernel<<<blocks, 256, 0, stream>>>(node_tokens, node_batch, active,
                                              sums, counts, NPW, N);

    finalize_kernel<<<B / 16, 32, 0, stream>>>(sums, counts, question, step_weight,
                                               step_counts, ln_gamma, ln_beta,
                                               (float*)d_out, max_steps);
}